// Model_61856118996995
// MI455X (gfx1250) — compile-verified
//
#include <hip/hip_runtime.h>
#include <math.h>

#define N_NODES 30000
#define NNZ_E   480000
#define N_ADJ   6
#define HDIM    128
#define DIN     64
#define ATTN_D  64

typedef __attribute__((ext_vector_type(2))) float v2f;
typedef __attribute__((ext_vector_type(8))) float v8f;

struct Coef { int k; float alpha; };

// ---------------------------------------------------------------------------
// Prep: compute the 16 (adjacency, scalar-weight) pairs on device.
// ---------------------------------------------------------------------------
__device__ float softmax_pick(const float* v, int n, int idx) {
    float m = v[0];
    for (int i = 1; i < n; ++i) m = fmaxf(m, v[i]);
    float s = 0.f;
    for (int i = 0; i < n; ++i) s += expf(v[i] - m);
    return expf(v[idx] - m) / s;
}

__global__ void prep_kernel(const float* as_seq0, const float* as_lseq0,
                            const float* as_res0, const float* as_lres0,
                            const float* as_seq1, const float* as_lseq1,
                            const float* as_res1, const float* as_lres1,
                            const int* i_seq0, const int* i_lseq0,
                            const int* i_res0, const int* i_lres0,
                            const int* i_seq1, const int* i_lseq1,
                            const int* i_res1, const int* i_lres1,
                            Coef* coefs) {
    if (threadIdx.x != 0 || blockIdx.x != 0) return;
    const int cext[4] = {0, 2, 4, 5};  // CSTR + [N_ADJS-1]
    int k, j;
    // cell0 (n_step = 4): 10 spmm terms
    k = i_seq0[0]; coefs[0]  = {k, softmax_pick(as_seq0 + 0 * 5, 5, k)};   // s1 <- s0
    k = i_seq0[1]; coefs[1]  = {k, softmax_pick(as_seq0 + 1 * 5, 5, k)};   // s2 <- s1
    k = i_res0[0]; coefs[2]  = {k, softmax_pick(as_res0 + 0 * 6, 6, k)};   // s2 <- s0
    k = i_seq0[2]; coefs[3]  = {k, softmax_pick(as_seq0 + 2 * 5, 5, k)};   // s3 <- s2
    k = i_res0[1]; coefs[4]  = {k, softmax_pick(as_res0 + 1 * 6, 6, k)};   // s3 <- s0
    k = i_res0[2]; coefs[5]  = {k, softmax_pick(as_res0 + 2 * 6, 6, k)};   // s3 <- s1
    j = i_lseq0[0]; coefs[6] = {cext[j], softmax_pick(as_lseq0, 3, j)};    // out <- s3
    j = i_lres0[0]; coefs[7] = {cext[j], softmax_pick(as_lres0 + 0 * 4, 4, j)}; // out <- s0
    j = i_lres0[1]; coefs[8] = {cext[j], softmax_pick(as_lres0 + 1 * 4, 4, j)}; // out <- s1
    j = i_lres0[2]; coefs[9] = {cext[j], softmax_pick(as_lres0 + 2 * 4, 4, j)}; // out <- s2
    // cell1 (n_step = 3): 6 spmm terms
    k = i_seq1[0]; coefs[10] = {k, softmax_pick(as_seq1 + 0 * 5, 5, k)};   // t1 <- t0
    k = i_seq1[1]; coefs[11] = {k, softmax_pick(as_seq1 + 1 * 5, 5, k)};   // t2 <- t1
    k = i_res1[0]; coefs[12] = {k, softmax_pick(as_res1 + 0 * 6, 6, k)};   // t2 <- t0
    j = i_lseq1[0]; coefs[13] = {cext[j], softmax_pick(as_lseq1, 3, j)};   // out1 <- t2
    j = i_lres1[0]; coefs[14] = {cext[j], softmax_pick(as_lres1 + 0 * 4, 4, j)}; // out1 <- t0
    j = i_lres1[1]; coefs[15] = {cext[j], softmax_pick(as_lres1 + 1 * 4, 4, j)}; // out1 <- t1
}

// ---------------------------------------------------------------------------
// Zero fill (float4 grid-stride) — used only for the CSR histogram counters.
// ---------------------------------------------------------------------------
__global__ void zero_kernel(float4* p, long n4) {
    long i = (long)blockIdx.x * blockDim.x + threadIdx.x;
    long stride = (long)gridDim.x * blockDim.x;
    float4 z = {0.f, 0.f, 0.f, 0.f};
    for (; i < n4; i += stride) p[i] = z;
}

// ---------------------------------------------------------------------------
// CSR build: histogram -> per-adjacency exclusive scan -> scatter -> row sort.
// Built once per launch; removes ALL atomics from the 16 SpMM passes and makes
// the float accumulation order deterministic (ascending original edge index,
// matching segment_sum's edge-order semantics).
// ---------------------------------------------------------------------------
__global__ void hist_kernel(const int* __restrict__ rows_all, int* __restrict__ counts) {
    int i = blockIdx.x * blockDim.x + threadIdx.x;
    if (i >= N_ADJ * NNZ_E) return;
    int k = i / NNZ_E;
    atomicAdd(&counts[k * N_NODES + rows_all[i]], 1);
}

__global__ void scan_kernel(const int* __restrict__ counts,
                            int* __restrict__ rowptr, int* __restrict__ cursor) {
    __shared__ int part[1024];
    const int k = blockIdx.x;              // one block per adjacency
    const int t = threadIdx.x;
    const int* cnt = counts + k * N_NODES;
    int* rp  = rowptr + k * (N_NODES + 1);
    int* cur = cursor + k * N_NODES;
    const int CH = (N_NODES + 1023) / 1024;   // 30 rows per thread
    const int beg = t * CH;

    int s = 0;
    for (int i = 0; i < CH; ++i) {
        int idx = beg + i;
        if (idx < N_NODES) s += cnt[idx];
    }
    part[t] = s;
    __syncthreads();
    // Hillis-Steele inclusive scan over 1024 partials
    for (int off = 1; off < 1024; off <<= 1) {
        int v = (t >= off) ? part[t - off] : 0;
        __syncthreads();
        part[t] += v;
        __syncthreads();
    }
    int run = (t == 0) ? 0 : part[t - 1];     // exclusive prefix of this chunk
    for (int i = 0; i < CH; ++i) {
        int idx = beg + i;
        if (idx < N_NODES) {
            rp[idx] = run;
            cur[idx] = run;
            run += cnt[idx];
        }
    }
    if (t == 1023) rp[N_NODES] = NNZ_E;
}

__global__ void scatter_kernel(const int* __restrict__ rows_all,
                               const int* __restrict__ cols_all,
                               const float* __restrict__ vals_all,
                               int* __restrict__ cursor,
                               int* __restrict__ cols_s, float* __restrict__ vals_s,
                               int* __restrict__ eidx_s) {
    int i = blockIdx.x * blockDim.x + threadIdx.x;
    if (i >= N_ADJ * NNZ_E) return;
    int k = i / NNZ_E;
    int e = i - k * NNZ_E;
    int r = rows_all[i];
    int pos = atomicAdd(&cursor[k * N_NODES + r], 1);   // pos in [0, NNZ) within adjacency
    long o = (long)k * NNZ_E + pos;
    cols_s[o] = cols_all[i];
    vals_s[o] = vals_all[i];
    eidx_s[o] = e;
}

// Stable within-row ordering (sort by original edge index) for determinism.
__global__ void sortrow_kernel(const int* __restrict__ rowptr,
                               int* __restrict__ cols_s, float* __restrict__ vals_s,
                               int* __restrict__ eidx_s) {
    int i = blockIdx.x * blockDim.x + threadIdx.x;
    if (i >= N_ADJ * N_NODES) return;
    int k = i / N_NODES, row = i - k * N_NODES;
    const int* rp = rowptr + k * (N_NODES + 1);
    int beg = rp[row], end = rp[row + 1];
    long off = (long)k * NNZ_E;
    for (int a = beg + 1; a < end; ++a) {
        int  ke = eidx_s[off + a];
        int  kc = cols_s[off + a];
        float kv = vals_s[off + a];
        int b = a - 1;
        while (b >= beg && eidx_s[off + b] > ke) {
            eidx_s[off + b + 1] = eidx_s[off + b];
            cols_s[off + b + 1] = cols_s[off + b];
            vals_s[off + b + 1] = vals_s[off + b];
            --b;
        }
        eidx_s[off + b + 1] = ke;
        cols_s[off + b + 1] = kc;
        vals_s[off + b + 1] = kv;
    }
}

// ---------------------------------------------------------------------------
// Per-type input projection: hid[n] = feats[n] @ W_in[type[n]] + b_in[type[n]]
// WMMA f32 16x16x4; 8 waves/block, block = 16 rows x 128 cols; all 3 type
// projections computed, per-row select. N_NODES % 16 == 0 -> EXEC all-1s.
// ---------------------------------------------------------------------------
__global__ void proj_kernel(const float* __restrict__ feats,
                            const int* __restrict__ types,
                            const float* __restrict__ W_in,
                            const float* __restrict__ b_in,
                            float* __restrict__ Y) {
    __shared__ float As[16][DIN];
    __shared__ int   Ty[16];
    const int base = blockIdx.x * 16;
    const int tid  = threadIdx.x;
    const int wave = tid >> 5;
    const int lane = tid & 31;

    for (int i = tid; i < 16 * DIN; i += 256) {
        int r = i >> 6, c = i & (DIN - 1);
        As[r][c] = feats[(long)(base + r) * DIN + c];
    }
    if (tid < 16) Ty[tid] = types[base + tid];
    __syncthreads();

    const int m    = lane & 15;
    const int hsel = lane >> 4;
    const int n0   = wave * 16;

    v8f a0 = {}, a1 = {}, a2 = {};
    for (int k0 = 0; k0 < DIN; k0 += 4) {
        v2f a, b0, b1, b2;
        int ka = k0 + 2 * hsel;
        a.x = As[m][ka];
        a.y = As[m][ka + 1];
        const float* w0 = W_in + 0 * DIN * HDIM;
        const float* w1 = W_in + 1 * DIN * HDIM;
        const float* w2 = W_in + 2 * DIN * HDIM;
        b0.x = w0[ka * HDIM + n0 + m];  b0.y = w0[(ka + 1) * HDIM + n0 + m];
        b1.x = w1[ka * HDIM + n0 + m];  b1.y = w1[(ka + 1) * HDIM + n0 + m];
        b2.x = w2[ka * HDIM + n0 + m];  b2.y = w2[(ka + 1) * HDIM + n0 + m];
        a0 = __builtin_amdgcn_wmma_f32_16x16x4_f32(false, a, false, b0, (short)0, a0, false, false);
        a1 = __builtin_amdgcn_wmma_f32_16x16x4_f32(false, a, false, b1, (short)0, a1, false, false);
        a2 = __builtin_amdgcn_wmma_f32_16x16x4_f32(false, a, false, b2, (short)0, a2, false, false);
    }
#pragma unroll
    for (int v = 0; v < 8; ++v) {
        int rloc = v + 8 * hsel;
        int t = Ty[rloc];
        float val = (t == 0) ? a0[v] : (t == 1) ? a1[v] : a2[v];
        val += b_in[t * HDIM + n0 + m];
        Y[(long)(base + rloc) * HDIM + n0 + m] = val;
    }
}

// ---------------------------------------------------------------------------
// Affine: Y = X @ W + b   (X: [N,128], W: [128,128] row-major)
// ---------------------------------------------------------------------------
__global__ void affine_kernel(const float* __restrict__ X,
                              const float* __restrict__ W,
                              const float* __restrict__ bias,
                              float* __restrict__ Y) {
    __shared__ float As[16][HDIM];
    const int base = blockIdx.x * 16;
    const int tid  = threadIdx.x;
    const int wave = tid >> 5;
    const int lane = tid & 31;

    for (int i = tid; i < 16 * HDIM; i += 256) {
        int r = i >> 7, c = i & (HDIM - 1);
        As[r][c] = X[(long)(base + r) * HDIM + c];
    }
    __syncthreads();

    const int m    = lane & 15;
    const int hsel = lane >> 4;
    const int n0   = wave * 16;

    const float bv = bias[n0 + m];
    v8f acc;
#pragma unroll
    for (int v = 0; v < 8; ++v) acc[v] = bv;

    for (int k0 = 0; k0 < HDIM; k0 += 4) {
        v2f a, b;
        int ka = k0 + 2 * hsel;
        a.x = As[m][ka];
        a.y = As[m][ka + 1];
        b.x = W[ka * HDIM + n0 + m];
        b.y = W[(ka + 1) * HDIM + n0 + m];
        acc = __builtin_amdgcn_wmma_f32_16x16x4_f32(false, a, false, b, (short)0, acc, false, false);
    }
#pragma unroll
    for (int v = 0; v < 8; ++v) {
        int rloc = v + 8 * hsel;
        Y[(long)(base + rloc) * HDIM + n0 + m] = acc[v];
    }
}

// ---------------------------------------------------------------------------
// Fused CSR SpMM: dst[row] = sum_{s<NSRC} alpha_s * (A_{k_s} @ src_s)[row]
// One wave per output row; lane covers 4 columns (float4 accumulator).
// Edge (col,val) pairs batch-loaded 32-at-a-time and shuffle-broadcast.
// Optionally fuses LayerNorm + exact GELU before the single store.
// ---------------------------------------------------------------------------
template <int NSRC, bool LNGELU>
__global__ void spmm_rows_kernel(const int* __restrict__ rowptr,
                                 const int* __restrict__ cols_s,
                                 const float* __restrict__ vals_s,
                                 const Coef* __restrict__ coefs, int4 opv,
                                 const float* __restrict__ x0, const float* __restrict__ x1,
                                 const float* __restrict__ x2, const float* __restrict__ x3,
                                 float* __restrict__ dst) {
    const int row  = (blockIdx.x * blockDim.x + threadIdx.x) >> 5;
    const int lane = threadIdx.x & 31;
    if (row >= N_NODES) return;

    const int opsA[4] = {opv.x, opv.y, opv.z, opv.w};
    const float* srcs[4] = {x0, x1, x2, x3};

    float4 acc = {0.f, 0.f, 0.f, 0.f};
#pragma unroll
    for (int s = 0; s < NSRC; ++s) {
        const Coef cf = coefs[opsA[s]];
        const int* rp = rowptr + cf.k * (N_NODES + 1);
        const int beg = rp[row], end = rp[row + 1];
        const long off = (long)cf.k * NNZ_E;
        const int*   cc = cols_s + off;
        const float* vv = vals_s + off;
        const float* x  = srcs[s];
        for (int base = beg; base < end; base += 32) {
            const int rem = end - base;
            int   cl = 0;
            float vl = 0.f;
            if (lane < rem) { cl = cc[base + lane]; vl = vv[base + lane]; }
            const int lim = rem < 32 ? rem : 32;
            for (int j = 0; j < lim; ++j) {
                const int   c = __shfl(cl, j, 32);
                const float w = cf.alpha * __shfl(vl, j, 32);
                const float4 xv = ((const float4*)(x + (long)c * HDIM))[lane];
                acc.x += w * xv.x;
                acc.y += w * xv.y;
                acc.z += w * xv.z;
                acc.w += w * xv.w;
            }
        }
    }

    if (LNGELU) {
        float ssum = acc.x + acc.y + acc.z + acc.w;
#pragma unroll
        for (int o = 16; o > 0; o >>= 1) ssum += __shfl_xor(ssum, o, 32);
        const float mu = ssum * (1.f / HDIM);
        const float dx = acc.x - mu, dy = acc.y - mu, dz = acc.z - mu, dw = acc.w - mu;
        float q = dx * dx + dy * dy + dz * dz + dw * dw;
#pragma unroll
        for (int o = 16; o > 0; o >>= 1) q += __shfl_xor(q, o, 32);
        const float inv = rsqrtf(q * (1.f / HDIM) + 1e-5f);
        const float kk = 0.70710678118654752440f;
        float t;
        t = dx * inv; acc.x = 0.5f * t * (1.f + erff(t * kk));
        t = dy * inv; acc.y = 0.5f * t * (1.f + erff(t * kk));
        t = dz * inv; acc.z = 0.5f * t * (1.f + erff(t * kk));
        t = dw * inv; acc.w = 0.5f * t * (1.f + erff(t * kk));
    }
    ((float4*)(dst + (long)row * HDIM))[lane] = acc;
}

// ---------------------------------------------------------------------------
// Attention + blend: a_i = tanh(h_i @ W1 + b1) @ W2 + b2 ; softmax over {0,1};
// out = p0*h0 + p1*h1. One 64-thread block per node.
// ---------------------------------------------------------------------------
__global__ void attn_blend_kernel(const float* __restrict__ h0,
                                  const float* __restrict__ h1,
                                  const float* __restrict__ W1,
                                  const float* __restrict__ b1,
                                  const float* __restrict__ W2,
                                  const float* __restrict__ b2,
                                  float* __restrict__ out) {
    __shared__ float sh0[HDIM], sh1[HDIM], red[ATTN_D];
    __shared__ float a0s, a1s;
    const int node = blockIdx.x;
    const int t = threadIdx.x;

    sh0[t]      = h0[(long)node * HDIM + t];
    sh0[t + 64] = h0[(long)node * HDIM + t + 64];
    sh1[t]      = h1[(long)node * HDIM + t];
    sh1[t + 64] = h1[(long)node * HDIM + t + 64];
    __syncthreads();

    float acc0 = b1[t], acc1 = b1[t];
    for (int d = 0; d < HDIM; ++d) {
        const float w = W1[d * ATTN_D + t];
        acc0 += sh0[d] * w;
        acc1 += sh1[d] * w;
    }
    const float w2 = W2[t];
    const float t0 = tanhf(acc0) * w2;
    const float t1 = tanhf(acc1) * w2;

    red[t] = t0; __syncthreads();
    for (int s = 32; s > 0; s >>= 1) { if (t < s) red[t] += red[t + s]; __syncthreads(); }
    if (t == 0) a0s = red[0] + b2[0];
    __syncthreads();
    red[t] = t1; __syncthreads();
    for (int s = 32; s > 0; s >>= 1) { if (t < s) red[t] += red[t + s]; __syncthreads(); }
    if (t == 0) a1s = red[0] + b2[0];
    __syncthreads();

    const float a0 = a0s, a1 = a1s;
    const float mx = fmaxf(a0, a1);
    const float e0 = expf(a0 - mx), e1 = expf(a1 - mx);
    const float invs = 1.f / (e0 + e1);
    const float p0 = e0 * invs, p1 = e1 * invs;

    out[(long)node * HDIM + t]      = p0 * sh0[t]      + p1 * sh1[t];
    out[(long)node * HDIM + t + 64] = p0 * sh0[t + 64] + p1 * sh1[t + 64];
}

// ---------------------------------------------------------------------------
// Launch
// ---------------------------------------------------------------------------
extern "C" void kernel_launch(void* const* d_in, const int* in_sizes, int n_in,
                              void* d_out, int out_size, void* d_ws, size_t ws_size,
                              hipStream_t stream) {
    const float* feats      = (const float*)d_in[0];
    const int*   node_types = (const int*)d_in[1];
    const int*   rows       = (const int*)d_in[2];
    const int*   cols       = (const int*)d_in[3];
    const float* vals       = (const float*)d_in[4];
    const float* W_in       = (const float*)d_in[5];
    const float* b_in       = (const float*)d_in[6];
    const float* affW0      = (const float*)d_in[7];
    const float* affb0      = (const float*)d_in[8];
    const float* affW1      = (const float*)d_in[9];
    const float* affb1      = (const float*)d_in[10];
    const float* as_seq0    = (const float*)d_in[11];
    const float* as_lseq0   = (const float*)d_in[12];
    const float* as_res0    = (const float*)d_in[13];
    const float* as_lres0   = (const float*)d_in[14];
    const float* as_seq1    = (const float*)d_in[15];
    const float* as_lseq1   = (const float*)d_in[16];
    const float* as_res1    = (const float*)d_in[17];
    const float* as_lres1   = (const float*)d_in[18];
    const int*   i_seq0     = (const int*)d_in[19];
    const int*   i_lseq0    = (const int*)d_in[20];
    const int*   i_res0     = (const int*)d_in[21];
    const int*   i_lres0    = (const int*)d_in[22];
    const int*   i_seq1     = (const int*)d_in[23];
    const int*   i_lseq1    = (const int*)d_in[24];
    const int*   i_res1     = (const int*)d_in[25];
    const int*   i_lres1    = (const int*)d_in[26];
    const float* attn_W1    = (const float*)d_in[27];
    const float* attn_b1    = (const float*)d_in[28];
    const float* attn_W2    = (const float*)d_in[29];
    const float* attn_b2    = (const float*)d_in[30];
    float* outp = (float*)d_out;

    const long NH = (long)N_NODES * HDIM;
    char* p = (char*)d_ws;
    auto alignup = [](size_t x) { return (x + 255) & ~(size_t)255; };

    size_t off = 0;
    Coef* coefs  = (Coef*)(p + off);  off = alignup(off + 16 * sizeof(Coef));
    int* counts  = (int*)(p + off);   off = alignup(off + (size_t)N_ADJ * N_NODES * 4);
    int* rowptr  = (int*)(p + off);   off = alignup(off + (size_t)N_ADJ * (N_NODES + 1) * 4);
    int* cursor  = (int*)(p + off);   off = alignup(off + (size_t)N_ADJ * N_NODES * 4);
    int* cols_s  = (int*)(p + off);   off = alignup(off + (size_t)N_ADJ * NNZ_E * 4);
    float* vals_s = (float*)(p + off); off = alignup(off + (size_t)N_ADJ * NNZ_E * 4);
    int* eidx_s  = (int*)(p + off);   off = alignup(off + (size_t)N_ADJ * NNZ_E * 4);
    float* hid   = (float*)(p + off); off += NH * 4;
    float* s0    = (float*)(p + off); off += NH * 4;   // cell0 x; reused as cell1 t0
    float* s1    = (float*)(p + off); off += NH * 4;   // reused as t1
    float* s2    = (float*)(p + off); off += NH * 4;   // reused as t2
    float* s3    = (float*)(p + off); off += NH * 4;   // reused as out1 -> h1
    float* out0  = (float*)(p + off); off += NH * 4;   // cell0 raw out -> h0

    const dim3 B256(256);
    const int gemm_blocks = N_NODES / 16;                    // 1875, exact
    const int edge_blocks = (N_ADJ * NNZ_E + 255) / 256;     // 11250
    const int row_blocks  = (N_ADJ * N_NODES + 255) / 256;   // 704
    const int spmm_blocks = (N_NODES + 7) / 8;               // wave per row
    const long cnt4       = (long)N_ADJ * N_NODES / 4;       // counts in float4 units

    // 0) coefficients (device-side index lookups + softmax)
    prep_kernel<<<1, 32, 0, stream>>>(as_seq0, as_lseq0, as_res0, as_lres0,
                                      as_seq1, as_lseq1, as_res1, as_lres1,
                                      i_seq0, i_lseq0, i_res0, i_lres0,
                                      i_seq1, i_lseq1, i_res1, i_lres1, coefs);
    // 1) CSR build for all 6 adjacencies
    zero_kernel<<<64, B256, 0, stream>>>((float4*)counts, cnt4);
    hist_kernel<<<edge_blocks, B256, 0, stream>>>(rows, counts);
    scan_kernel<<<N_ADJ, 1024, 0, stream>>>(counts, rowptr, cursor);
    scatter_kernel<<<edge_blocks, B256, 0, stream>>>(rows, cols, vals, cursor,
                                                     cols_s, vals_s, eidx_s);
    sortrow_kernel<<<row_blocks, B256, 0, stream>>>(rowptr, cols_s, vals_s, eidx_s);
    // 2) hid = per-type projection (WMMA)
    proj_kernel<<<gemm_blocks, B256, 0, stream>>>(feats, node_types, W_in, b_in, hid);
    // 3) cell0: s0 = hid @ affW0 + affb0 (WMMA)
    affine_kernel<<<gemm_blocks, B256, 0, stream>>>(hid, affW0, affb0, s0);
    // 4) cell0 fused SpMM DAG (gather, no atomics)
    spmm_rows_kernel<1, false><<<spmm_blocks, B256, 0, stream>>>(
        rowptr, cols_s, vals_s, coefs, make_int4(0, 0, 0, 0), s0, s0, s0, s0, s1);
    spmm_rows_kernel<2, false><<<spmm_blocks, B256, 0, stream>>>(
        rowptr, cols_s, vals_s, coefs, make_int4(1, 2, 0, 0), s1, s0, s0, s0, s2);
    spmm_rows_kernel<3, false><<<spmm_blocks, B256, 0, stream>>>(
        rowptr, cols_s, vals_s, coefs, make_int4(3, 4, 5, 0), s2, s0, s1, s0, s3);
    spmm_rows_kernel<4, true><<<spmm_blocks, B256, 0, stream>>>(
        rowptr, cols_s, vals_s, coefs, make_int4(6, 7, 8, 9), s3, s0, s1, s2, out0);
    // out0 now holds h0 = gelu(layernorm(raw))
    // 5) cell1: t0 = hid @ affW1 + affb1 (reuse s0)
    affine_kernel<<<gemm_blocks, B256, 0, stream>>>(hid, affW1, affb1, s0);
    spmm_rows_kernel<1, false><<<spmm_blocks, B256, 0, stream>>>(
        rowptr, cols_s, vals_s, coefs, make_int4(10, 0, 0, 0), s0, s0, s0, s0, s1);
    spmm_rows_kernel<2, false><<<spmm_blocks, B256, 0, stream>>>(
        rowptr, cols_s, vals_s, coefs, make_int4(11, 12, 0, 0), s1, s0, s0, s0, s2);
    spmm_rows_kernel<3, true><<<spmm_blocks, B256, 0, stream>>>(
        rowptr, cols_s, vals_s, coefs, make_int4(13, 14, 15, 0), s2, s0, s1, s0, s3);
    // s3 now holds h1
    // 6) attention softmax blend -> d_out
    attn_blend_kernel<<<N_NODES, 64, 0, stream>>>(out0, s3, attn_W1, attn_b1,
                                                  attn_W2, attn_b2, outp);
}